// Subgraph_33809982554186
// MI455X (gfx1250) — compile-verified
//
#include <hip/hip_runtime.h>
#include <hip/hip_bf16.h>

// ---------------- problem constants ----------------
#define NG   256      // graphs
#define NN   512      // nodes per graph
#define NE   8192     // edges per graph
#define FIN  128
#define DD1  256
#define DD2  256
#define HH1  64
#define CLSH 128

typedef __attribute__((ext_vector_type(2))) float v2f;
typedef __attribute__((ext_vector_type(8))) float v8f;

// ---------------- generic f32 WMMA GEMM: Y[g] = epi(X[g] @ W + bias) --------
// epi: 0 = none, 1 = +bias,relu, 2 = +bias,tanh
// Each wave computes a 64x16 output strip: 4 stacked 16x16 tiles sharing one
// B fragment per k-step (4x B reuse, 4 independent accumulator chains).
// 8 waves (256 threads) per block. Requires M % 64 == 0, N % 16 == 0.
__global__ void gemm_wmma_f32(const float* __restrict__ X,
                              const float* __restrict__ W,
                              const float* __restrict__ bias,
                              float* __restrict__ Y,
                              int M, int K, int N,
                              long long sx, long long sy, int epi) {
  const int g = blockIdx.y;
  const float* Xg = X + (long long)g * sx;
  float* Yg = Y + (long long)g * sy;

  const int wave = threadIdx.x >> 5;
  const int lane = threadIdx.x & 31;
  const int ntile = N >> 4;                 // 16-wide N tiles
  const int tile = blockIdx.x * 8 + wave;   // strip index over (M/64)*(N/16)
  const int tm = tile / ntile;
  const int tn = tile - tm * ntile;
  const int m0 = tm << 6, n0 = tn << 4;

  const int r  = lane & 15;          // A row-in-tile / B col-in-tile
  const int kg = (lane >> 4) << 1;   // lanes 16-31 hold K+2,K+3

  v8f acc0 = {}, acc1 = {}, acc2 = {}, acc3 = {};
  const float* ap0 = Xg + (long long)(m0 + r) * K;
  const float* ap1 = ap0 + (long long)16 * K;
  const float* ap2 = ap0 + (long long)32 * K;
  const float* ap3 = ap0 + (long long)48 * K;
  const float* bp  = W + n0 + r;

  for (int k0 = 0; k0 < K; k0 += 4) {
    const int ka = k0 + kg;
    v2f b;
    b.x = bp[(long long)ka * N];
    b.y = bp[(long long)(ka + 1) * N];
    v2f a0, a1, a2, a3;
    a0.x = ap0[ka]; a0.y = ap0[ka + 1];
    a1.x = ap1[ka]; a1.y = ap1[ka + 1];
    a2.x = ap2[ka]; a2.y = ap2[ka + 1];
    a3.x = ap3[ka]; a3.y = ap3[ka + 1];
    acc0 = __builtin_amdgcn_wmma_f32_16x16x4_f32(false, a0, false, b, (short)0, acc0, false, false);
    acc1 = __builtin_amdgcn_wmma_f32_16x16x4_f32(false, a1, false, b, (short)0, acc1, false, false);
    acc2 = __builtin_amdgcn_wmma_f32_16x16x4_f32(false, a2, false, b, (short)0, acc2, false, false);
    acc3 = __builtin_amdgcn_wmma_f32_16x16x4_f32(false, a3, false, b, (short)0, acc3, false, false);
  }

  const int row_base = m0 + ((lane >> 4) << 3);  // + i covers 16 rows per tile
  const int col = n0 + (lane & 15);
  const float bv = (epi != 0) ? bias[col] : 0.0f;

  v8f accs[4] = {acc0, acc1, acc2, acc3};
#pragma unroll
  for (int tsub = 0; tsub < 4; ++tsub) {
#pragma unroll
    for (int i = 0; i < 8; ++i) {
      float v = accs[tsub][i] + bv;
      if (epi == 1) v = fmaxf(v, 0.0f);
      else if (epi == 2) v = tanhf(v);
      Yg[(long long)(row_base + (tsub << 4) + i) * N + col] = v;
    }
  }
}

// ---------------- degree / dinv ----------------
__global__ void degree_kernel(const int* __restrict__ edges, float* __restrict__ dinv) {
  __shared__ float sdeg[NN];
  const int g = blockIdx.x, t = threadIdx.x;
  sdeg[t] = 0.f; sdeg[t + 256] = 0.f;
  __syncthreads();
  const int* eg = edges + (size_t)g * 2 * NE;
  for (int e = t; e < NE; e += 256) atomicAdd(&sdeg[eg[NE + e]], 1.0f);
  __syncthreads();
  float* dv = dinv + (size_t)g * NN;
  dv[t]       = rsqrtf(sdeg[t] + 1.0f);        // +1 for self loop
  dv[t + 256] = rsqrtf(sdeg[t + 256] + 1.0f);
}

// ---------------- edge scatter: out[d] += src[s] * dinv[s]*dinv[d] ----------
// one wave per (edge or self-loop); lanes cover D columns.
__global__ void scatter_kernel(const float* __restrict__ src,
                               const int* __restrict__ edges,
                               const float* __restrict__ dinv,
                               float* __restrict__ out, int D) {
  const int g = blockIdx.y;
  const int wave = threadIdx.x >> 5;
  const int lane = threadIdx.x & 31;
  const int w = blockIdx.x * 8 + wave;
  if (w >= NE + NN) return;
  const int* eg = edges + (size_t)g * 2 * NE;
  const float* dv = dinv + (size_t)g * NN;
  int s, d; float nrm;
  if (w < NE) { s = eg[w]; d = eg[NE + w]; nrm = dv[s] * dv[d]; }
  else        { s = d = w - NE;            nrm = dv[s] * dv[s]; }
  const float* sr = src + ((size_t)g * NN + s) * D;
  float* ot = out + ((size_t)g * NN + d) * D;
  for (int j = lane; j < D; j += 32) atomicAdd(&ot[j], sr[j] * nrm);
}

// ---------------- elementwise: Y = act(Y + bias[col]) ----------------
__global__ void bias_act_kernel(float* __restrict__ Y, const float* __restrict__ bias,
                                long long total, int D, int relu) {
  long long i = blockIdx.x * (long long)blockDim.x + threadIdx.x;
  const long long st = (long long)gridDim.x * blockDim.x;
  for (; i < total; i += st) {
    float v = Y[i] + bias[(int)(i % D)];
    Y[i] = relu ? fmaxf(v, 0.0f) : v;
  }
}

__global__ void zero_kernel(float* __restrict__ p, long long n) {
  long long i = blockIdx.x * (long long)blockDim.x + threadIdx.x;
  const long long st = (long long)gridDim.x * blockDim.x;
  for (; i < n; i += st) p[i] = 0.0f;
}

// ---------------- per-node 2-way softmax assignment ----------------
__global__ void assign_kernel(const float* __restrict__ a1, const float* __restrict__ W4,
                              const float* __restrict__ b4, float* __restrict__ assign) {
  const long long idx = blockIdx.x * (long long)blockDim.x + threadIdx.x; // g*NN + n
  if (idx >= (long long)NG * NN) return;
  const float* a = a1 + idx * HH1;
  float z0 = b4[0], z1 = b4[1];
#pragma unroll 8
  for (int h = 0; h < HH1; ++h) { float v = a[h]; z0 += v * W4[2*h]; z1 += v * W4[2*h+1]; }
  float m = fmaxf(z0, z1);
  float e0 = expf(z0 - m), e1 = expf(z1 - m);
  float inv = 1.0f / (e0 + e1);
  assign[idx*2]     = e0 * inv;
  assign[idx*2 + 1] = e1 * inv;
}

// ---------------- per-graph pooling: group, new_adj, penalty, outputs -------
__global__ void pool_kernel(const float* __restrict__ h2, const float* __restrict__ assign,
                            const int* __restrict__ edges, float* __restrict__ pen,
                            float* __restrict__ out) {
  __shared__ float sa[2 * NN];
  __shared__ float adj[2 * NN];
  __shared__ float red[4];
  const int g = blockIdx.x, t = threadIdx.x;
  const float* ag = assign + (size_t)g * 2 * NN;
  for (int i = t; i < 2 * NN; i += 256) { sa[i] = ag[i]; adj[i] = 0.0f; }
  if (t < 4) red[t] = 0.0f;
  __syncthreads();

  // group[k, t] = sum_n assign[n,k] * h2[n,t]   (t = column 0..255)
  const float* h2g = h2 + (size_t)g * NN * DD2;
  float g0 = 0.0f, g1 = 0.0f;
  for (int n = 0; n < NN; ++n) {
    float h = h2g[(size_t)n * DD2 + t];
    g0 += sa[2*n]     * h;
    g1 += sa[2*n + 1] * h;
  }

  // (A @ assign)[src] += assign[dst]  (duplicate edges accumulate, matching .at[].add)
  const int* eg = edges + (size_t)g * 2 * NE;
  for (int e = t; e < NE; e += 256) {
    int s = eg[e], d = eg[NE + e];
    atomicAdd(&adj[2*s],     sa[2*d]);
    atomicAdd(&adj[2*s + 1], sa[2*d + 1]);
  }
  __syncthreads();

  // new_adj = assign^T @ (A @ assign)  -> 2x2 reduction
  float c00 = 0, c01 = 0, c10 = 0, c11 = 0;
  for (int n = t; n < NN; n += 256) {
    float a0 = sa[2*n], a1 = sa[2*n + 1], b0 = adj[2*n], b1 = adj[2*n + 1];
    c00 += a0 * b0; c01 += a0 * b1; c10 += a1 * b0; c11 += a1 * b1;
  }
  atomicAdd(&red[0], c00); atomicAdd(&red[1], c01);
  atomicAdd(&red[2], c10); atomicAdd(&red[3], c11);
  __syncthreads();
  if (t == 0) {
    float r0 = fmaxf(fabsf(red[0]) + fabsf(red[1]), 1e-12f);
    float r1 = fmaxf(fabsf(red[2]) + fabsf(red[3]), 1e-12f);
    float d0 = red[0] / r0 - 1.0f;
    float d1 = red[3] / r1 - 1.0f;
    pen[g] = 0.5f * (d0 * d0 + d1 * d1);
  }

  float pos = fminf(fmaxf(g0, -100.0f), 100.0f);
  float neg = fminf(fmaxf(g1, -100.0f), 100.0f);
  out[(size_t)g * DD2 + t]              = 0.5f * (g0 + g1);   // emb
  out[(size_t)(NG  + g) * DD2 + t]      = pos;                // pos
  out[(size_t)(2*NG + g) * DD2 + t]     = neg;                // neg
}

// ---------------- final: pred = relu(hidden@Wc2+bc2), cls_loss + penalty ----
__global__ void final_kernel(const float* __restrict__ hidden, const float* __restrict__ Wc2,
                             const float* __restrict__ bc2, const float* __restrict__ labels,
                             const float* __restrict__ pen, float* __restrict__ out2) {
  __shared__ float s[2 * NG];            // 512
  const int t = threadIdx.x;             // 0..511 rows of concat(emb,pos)
  const float* h = hidden + (size_t)t * CLSH;
  float z = bc2[0];
#pragma unroll 8
  for (int i = 0; i < CLSH; ++i) z += h[i] * Wc2[i];
  float pred = fmaxf(z, 0.0f);
  float lab = labels[t & (NG - 1)];
  float e = pred - lab;
  s[t] = e * e;
  __syncthreads();
  for (int off = NG; off > 0; off >>= 1) { if (t < off) s[t] += s[t + off]; __syncthreads(); }
  if (t == 0) out2[0] = s[0] / (2.0f * NG);
  __syncthreads();
  s[t] = (t < NG) ? pen[t] : 0.0f;
  __syncthreads();
  for (int off = NG; off > 0; off >>= 1) { if (t < off) s[t] += s[t + off]; __syncthreads(); }
  if (t == 0) out2[1] = 5.0f * (s[0] / (float)NG);
}

// ---------------- host side ----------------
extern "C" void kernel_launch(void* const* d_in, const int* in_sizes, int n_in,
                              void* d_out, int out_size, void* d_ws, size_t ws_size,
                              hipStream_t stream) {
  (void)in_sizes; (void)n_in; (void)out_size; (void)ws_size;
  const float* features = (const float*)d_in[0];
  const int*   edges    = (const int*)  d_in[1];
  const float* labels   = (const float*)d_in[2];
  const float* W1 = (const float*)d_in[3];  const float* b1 = (const float*)d_in[4];
  const float* W2 = (const float*)d_in[5];  const float* b2 = (const float*)d_in[6];
  const float* W3 = (const float*)d_in[7];  const float* b3 = (const float*)d_in[8];
  const float* W4 = (const float*)d_in[9];  const float* b4 = (const float*)d_in[10];
  const float* Wc1 = (const float*)d_in[11]; const float* bc1 = (const float*)d_in[12];
  const float* Wc2 = (const float*)d_in[13]; const float* bc2 = (const float*)d_in[14];
  float* out = (float*)d_out;

  // workspace layout (floats)
  const long long BIG = (long long)NG * NN * DD2;       // 33,554,432
  float* ws     = (float*)d_ws;
  float* dinv   = ws;                                   // NG*NN
  float* assign = ws + (long long)NG * NN;              // NG*NN*2
  float* pen    = assign + (long long)NG * NN * 2;      // NG
  float* hidden = pen + NG;                             // 2*NG*CLSH = 65536
  float* bufA   = hidden + (long long)2 * NG * CLSH;    // BIG
  float* bufB   = bufA + BIG;                           // BIG

  const dim3 blk(256);
  const int scat_blocks = (NE + NN) / 8;                // 1088

  // strips per GEMM: (M/64)*(N/16)/8 blocks in x
  // 1) degrees -> dinv
  degree_kernel<<<dim3(NG), blk, 0, stream>>>(edges, dinv);

  // 2) xw = feat @ W1 -> A   (M=512,K=128,N=256: 8*16/8 = 16 blocks)
  gemm_wmma_f32<<<dim3(16, NG), blk, 0, stream>>>(features, W1, nullptr, bufA,
      NN, FIN, DD1, (long long)NN * FIN, (long long)NN * DD1, 0);

  // 3) scatter -> B ; h1 = relu(B + b1)
  zero_kernel<<<dim3(4096), blk, 0, stream>>>(bufB, BIG);
  scatter_kernel<<<dim3(scat_blocks, NG), blk, 0, stream>>>(bufA, edges, dinv, bufB, DD1);
  bias_act_kernel<<<dim3(4096), blk, 0, stream>>>(bufB, b1, BIG, DD1, 1);

  // 4) h1w = h1 @ W2 -> A ; scatter -> B ; h2 = B + b2
  gemm_wmma_f32<<<dim3(16, NG), blk, 0, stream>>>(bufB, W2, nullptr, bufA,
      NN, DD1, DD2, (long long)NN * DD1, (long long)NN * DD2, 0);
  zero_kernel<<<dim3(4096), blk, 0, stream>>>(bufB, BIG);
  scatter_kernel<<<dim3(scat_blocks, NG), blk, 0, stream>>>(bufA, edges, dinv, bufB, DD2);
  bias_act_kernel<<<dim3(4096), blk, 0, stream>>>(bufB, b2, BIG, DD2, 0);

  // 5) a1 = tanh(h2 @ W3 + b3) -> A  (N=64: 8*4/8 = 4 blocks) ; assignment
  gemm_wmma_f32<<<dim3(4, NG), blk, 0, stream>>>(bufB, W3, b3, bufA,
      NN, DD2, HH1, (long long)NN * DD2, (long long)NN * HH1, 2);
  assign_kernel<<<dim3((NG * NN) / 256), blk, 0, stream>>>(bufA, W4, b4, assign);

  // 6) pooling: emb/pos/neg -> d_out, per-graph penalty -> pen
  pool_kernel<<<dim3(NG), blk, 0, stream>>>(bufB, assign, edges, pen, out);

  // 7) classifier: data = d_out[0 : 2G*D2] (= concat(emb,pos) rows)
  //    M=512,K=256,N=128: 8*8/8 = 8 blocks
  gemm_wmma_f32<<<dim3(8, 1), blk, 0, stream>>>(out, Wc1, bc1, hidden,
      2 * NG, DD2, CLSH, 0, 0, 1);
  final_kernel<<<dim3(1), dim3(512), 0, stream>>>(hidden, Wc2, bc2, labels, pen,
      out + (long long)3 * NG * DD2);
}